// Embed_13580686590195
// MI455X (gfx1250) — compile-verified
//
#include <hip/hip_runtime.h>

// Embed: out[b,p,:] = W[argmax(onehot[b,p,:]), :] * sqrt(768) + bias
// One-hot einsum degenerates to an index-scan + row gather (pure bandwidth,
// ~850 MB total -> ~37us at 23.3 TB/s on MI455X). No matrix ops survive the
// algebraic simplification. CDNA5-specific paths used:
//   - global_load_b128 th:TH_LOAD_NT for the 823 MB one-hot stream
//   - global_prefetch_b8 (clamped; WGP-scope prefetch is non-speculative)
//   - global_load_async_to_lds_b128 + s_wait_asynccnt for gather-phase
//     tile staging (W row + bias -> LDS).

#define D_VOCAB 50257
#define D_MODEL 768
#define NROWS   4096                       // BATCH * POS = 2 * 2048
#define NVEC    ((NROWS * D_VOCAB) / 4)    // 51,463,168 16-byte chunks (exact)

typedef unsigned int u4 __attribute__((ext_vector_type(4)));
typedef float        f4 __attribute__((ext_vector_type(4)));

#if defined(__HIP_DEVICE_COMPILE__) && \
    __has_builtin(__builtin_amdgcn_global_load_async_to_lds_b128)
#define HAS_ASYNC_LDS 1
// Builtin signature (from clang diagnostic): param 0 is
//   __attribute__((vector_size(16))) int  addrspace(1)*   (global src)
// and by the ISA, param 1 is the LDS destination (addrspace(3)).
typedef int i4g __attribute__((vector_size(4 * sizeof(int))));
typedef __attribute__((address_space(1))) i4g gbl_i4;
typedef __attribute__((address_space(3))) i4g lds_i4;
// generic->AS1: numerically identity on AMDGPU.
// generic->AS3: generic LDS addresses carry the DS offset in the low 32 bits.
__device__ __forceinline__ gbl_i4* to_gbl(const void* p) {
  return (gbl_i4*)(unsigned long long)p;
}
__device__ __forceinline__ lds_i4* to_lds(const void* p) {
  return (lds_i4*)(unsigned int)(unsigned long long)p;
}
#endif

__device__ __forceinline__ void record_hit(unsigned g, int* __restrict__ ids,
                                           int ids_stride) {
  // g = flat float index of a nonzero one-hot element
  unsigned row = g / (unsigned)D_VOCAB;            // -> umulhi by compiler
  unsigned col = g - row * (unsigned)D_VOCAB;
  ids[(size_t)row * (size_t)ids_stride] = (int)col;
}

// Phase 1: stream the whole one-hot tensor as aligned 16B chunks (flat view:
// per-row pointers are NOT 16B aligned since 50257 is odd; the flat base is).
// Non-temporal loads keep the 823 MB stream from polluting L2.
__global__ __launch_bounds__(256) void onehot_scan_kernel(
    const u4* __restrict__ tok, int* __restrict__ ids, int ids_stride)
{
  int i = blockIdx.x * 256 + threadIdx.x;
  const int stride = gridDim.x * 256;
  const int nvec = NVEC;

  for (; i < nvec; i += stride) {
    u4 v = __builtin_nontemporal_load(&tok[i]);   // global_load_b128 th:NT

    // gfx1250 global_prefetch_b8, 2 grid-strides ahead (~66 MB, fits L2).
    int pf = i + 2 * stride;
    if (pf >= nvec) pf = nvec - 1;
    __builtin_prefetch((const void*)&tok[pf], 0, 0);

    // one-hot: nonzero bit pattern == 1.0f; integer OR-reduce is cheapest
    if (v.x | v.y | v.z | v.w) {
      unsigned f = (unsigned)i * 4u;
      // A chunk can straddle a row boundary and (rarely) hold two hits:
      // check all four elements independently.
      if (v.x) record_hit(f + 0u, ids, ids_stride);
      if (v.y) record_hit(f + 1u, ids, ids_stride);
      if (v.z) record_hit(f + 2u, ids, ids_stride);
      if (v.w) record_hit(f + 3u, ids, ids_stride);
    }
  }
}

// Phase 2: one 192-thread block (6 wave32s) per row; stage W[id] row + bias
// into LDS with the CDNA5 async global->LDS mover (one b128 per lane,
// ASYNCcnt-tracked), then out = W[id]*sqrt(768) + bias. Mul-then-add rounding
// (no FMA contraction) to match the reference bit-for-bit.
__global__ __launch_bounds__(192) void embed_gather_kernel(
    const f4* __restrict__ W, const f4* __restrict__ bias,
    const int* __restrict__ ids, int ids_stride, f4* __restrict__ out)
{
  const int row = blockIdx.x;
  const int tid = threadIdx.x;                    // 0..191 (192*4 = 768)
  __shared__ int sid;
#if HAS_ASYNC_LDS
  __shared__ f4 s_tile[2][D_MODEL / 4];           // [0]=W row, [1]=bias
#endif
  if (tid == 0) sid = ids[(size_t)row * (size_t)ids_stride];
  __syncthreads();          // also orders the ids-in-out fallback read/write
  const int id = sid;

  const f4* wrow = W + (size_t)id * (D_MODEL / 4);
  f4 w, b;
#if HAS_ASYNC_LDS
  // Async copy: each lane moves its own 16B chunk of the W row and of bias.
  __builtin_amdgcn_global_load_async_to_lds_b128(
      to_gbl(&wrow[tid]), to_lds(&s_tile[0][tid]), 0, 0);
  __builtin_amdgcn_global_load_async_to_lds_b128(
      to_gbl(&bias[tid]), to_lds(&s_tile[1][tid]), 0, 0);
#if __has_builtin(__builtin_amdgcn_s_wait_asynccnt)
  __builtin_amdgcn_s_wait_asynccnt(0);
#else
  asm volatile("s_wait_asynccnt 0x0" ::: "memory");
#endif
  __syncthreads();
  w = s_tile[0][tid];
  b = s_tile[1][tid];
#else
  w = wrow[tid];
  b = bias[tid];
#endif

  const float scale = 27.712812921102035f;        // sqrt(768)
  f4 o;
  o.x = __fadd_rn(__fmul_rn(w.x, scale), b.x);
  o.y = __fadd_rn(__fmul_rn(w.y, scale), b.y);
  o.z = __fadd_rn(__fmul_rn(w.z, scale), b.z);
  o.w = __fadd_rn(__fmul_rn(w.w, scale), b.w);

  out[(size_t)row * (D_MODEL / 4) + tid] = o;
}

extern "C" void kernel_launch(void* const* d_in, const int* in_sizes, int n_in,
                              void* d_out, int out_size, void* d_ws, size_t ws_size,
                              hipStream_t stream) {
  const u4* tok  = (const u4*)d_in[0];   // one-hot [4096, 50257] f32 (flat)
  const f4* W    = (const f4*)d_in[1];   // [50257, 768] f32
  const f4* bias = (const f4*)d_in[2];   // [768] f32
  f4*       out  = (f4*)d_out;           // [4096, 768] f32

  // ids scratch: 16 KB in d_ws; deterministic fallback stashes each row's id
  // in out[row*768] (read + block barrier before that slot is overwritten).
  int* ids;
  int  ids_stride;
  if (ws_size >= (size_t)NROWS * sizeof(int)) {
    ids = (int*)d_ws;
    ids_stride = 1;
  } else {
    ids = (int*)d_out;
    ids_stride = D_MODEL;
  }

  onehot_scan_kernel<<<8192, 256, 0, stream>>>(tok, ids, ids_stride);
  embed_gather_kernel<<<NROWS, 192, 0, stream>>>(W, bias, ids, ids_stride, out);
}